// AdaptiveReLU_85624468013527
// MI455X (gfx1250) — compile-verified
//
#include <hip/hip_runtime.h>

typedef __attribute__((ext_vector_type(2))) float v2f;
typedef __attribute__((ext_vector_type(4))) float v4f;
typedef __attribute__((ext_vector_type(8))) float v8f;

// Order-preserving float<->uint mapping so segment min/max can use native
// global_atomic_min_u32 / global_atomic_max_u32 (fast, deterministic).
__device__ __forceinline__ unsigned fenc(float f) {
    unsigned u = __float_as_uint(f);
    return (u & 0x80000000u) ? ~u : (u | 0x80000000u);
}
__device__ __forceinline__ float fdec(unsigned u) {
    u = (u & 0x80000000u) ? (u & 0x7FFFFFFFu) : ~u;
    return __uint_as_float(u);
}

#define ENC_PINF 0xFF800000u  // fenc(+inf): identity for segment-min
#define ENC_NINF 0x007FFFFFu  // fenc(-inf): identity for segment-max

// ---------------------------------------------------------------- init
__global__ void ar_init_kernel(unsigned* __restrict__ mn, unsigned* __restrict__ mx,
                               float* __restrict__ sums, float* __restrict__ relu,
                               float* __restrict__ counts, int SF, int S) {
    int stride = gridDim.x * blockDim.x;
    for (int i = blockIdx.x * blockDim.x + threadIdx.x; i < SF; i += stride) {
        mn[i]   = ENC_PINF;
        mx[i]   = ENC_NINF;
        sums[i] = 0.0f;
        relu[i] = 0.0f;
        if (i < S) counts[i] = 0.0f;
    }
}

// ---------------------------------------------------------------- pass 1
// One thread per (instance, 4-feature group): scatter min/max/sum/count.
// x is streamed once with NO reuse -> non-temporal load, so the 64 MB of
// L2-resident stat arrays (atomic targets) are not evicted by 512 MB of x.
// SHIFT >= 0: G == 1<<SHIFT (fast shift/mask path); SHIFT < 0: runtime div.
template <int SHIFT>
__global__ void ar_scatter_stats_kernel(const float* __restrict__ x,
                                        const int* __restrict__ idx,
                                        unsigned* __restrict__ mn, unsigned* __restrict__ mx,
                                        float* __restrict__ sums, float* __restrict__ counts,
                                        int N, int F, int G) {
    int g = blockIdx.x * blockDim.x + threadIdx.x;
    int total = N * G;
    if (g >= total) return;
    int i, fg;
    if constexpr (SHIFT >= 0) {
        i  = g >> SHIFT;
        fg = (g & ((1 << SHIFT) - 1)) << 2;
    } else {
        i  = g / G;
        fg = (g - i * G) << 2;
    }
    int s = idx[i];

    v4f xv = __builtin_nontemporal_load((const v4f*)(x + (size_t)i * F + fg));

    size_t base = (size_t)s * F + fg;
#pragma unroll
    for (int k = 0; k < 4; ++k) {
        float v    = xv[k];
        unsigned u = fenc(v);
        atomicMin(mn + base + k, u);      // global_atomic_min_u32 (L2-resident)
        atomicMax(mx + base + k, u);      // global_atomic_max_u32
        atomicAdd(sums + base + k, v);    // global_atomic_add_f32
    }
    if (fg == 0) atomicAdd(counts + s, 1.0f);
}

// ---------------------------------------------------------------- pass 2
// bias = t*max + (1-t)*min (gather, L2-resident), scatter-add relu(x - bias).
template <int SHIFT>
__global__ void ar_relu_scatter_kernel(const float* __restrict__ x,
                                       const int* __restrict__ idx,
                                       const float* __restrict__ t,
                                       const unsigned* __restrict__ mn,
                                       const unsigned* __restrict__ mx,
                                       float* __restrict__ relu,
                                       int N, int F, int G) {
    int g = blockIdx.x * blockDim.x + threadIdx.x;
    int total = N * G;
    if (g >= total) return;
    int i, fg;
    if constexpr (SHIFT >= 0) {
        i  = g >> SHIFT;
        fg = (g & ((1 << SHIFT) - 1)) << 2;
    } else {
        i  = g / G;
        fg = (g - i * G) << 2;
    }
    int s = idx[i];

    v4f xv = __builtin_nontemporal_load((const v4f*)(x + (size_t)i * F + fg));

    size_t base = (size_t)s * F + fg;
#pragma unroll
    for (int k = 0; k < 4; ++k) {
        float tt = t[fg + k];
        tt = fminf(fmaxf(tt, 0.0f), 1.0f);                 // clamp_convex_weight
        float bias = tt * fdec(mx[base + k]) + (1.0f - tt) * fdec(mn[base + k]);
        float r = xv[k] - bias;
        if (r > 0.0f) atomicAdd(relu + base + k, r);       // skip zero adds
    }
}

// ---------------------------------------------------------------- finalize (WMMA)
// out[e] = W0*count + W1*mn + W2*mx + W3*relu + W4*sums for 16 consecutive e
// per wave via one V_WMMA_F32_16X16X4_F32:
//   A(16x4)[m,k] = {mn,mx,relu,sums}[base+m],  B(4x16)[k,n] = W[k+1],
//   C = W0*count (all 16 elems share one segment since 16 | F).
// A layout: lanes 0-15 hold K=0,1 (VGPR0,1); lanes 16-31 hold K=2,3.
// B layout: lanes 0-15 hold rows K=0,1; lanes 16-31 rows K=2,3 (const over N).
// Stats are dead after this read (NT load); out is never re-read (NT store).
__global__ void ar_finalize_wmma_kernel(const unsigned* __restrict__ mn,
                                        const unsigned* __restrict__ mx,
                                        const float* __restrict__ sums,
                                        const float* __restrict__ relu,
                                        const float* __restrict__ counts,
                                        const float* __restrict__ W,
                                        float* __restrict__ out,
                                        int ntiles, int F) {
    int wid  = (blockIdx.x * blockDim.x + threadIdx.x) >> 5;
    int lane = threadIdx.x & 31;
    if (wid < ntiles) {                       // wave-uniform: EXEC stays all-1s
        int base = wid << 4;
        int e    = base + (lane & 15);
        int s    = base / F;

        v2f a, b;
        if (lane < 16) {
            a.x = fdec(__builtin_nontemporal_load(mn + e));
            a.y = fdec(__builtin_nontemporal_load(mx + e));
            b.x = W[1];  b.y = W[2];
        } else {
            a.x = __builtin_nontemporal_load(relu + e);
            a.y = __builtin_nontemporal_load(sums + e);
            b.x = W[3];  b.y = W[4];
        }
        float c0 = W[0] * counts[s];
        v8f c = {c0, c0, c0, c0, c0, c0, c0, c0};

        v8f d = __builtin_amdgcn_wmma_f32_16x16x4_f32(
            /*neg_a=*/false, a, /*neg_b=*/false, b,
            /*c_mod=*/(short)0, c, /*reuse_a=*/false, /*reuse_b=*/false);

        // D[m,n] identical over n; lane 0 holds M=0..7, lane 16 holds M=8..15.
        if (lane == 0) {
            __builtin_nontemporal_store((v4f){d[0], d[1], d[2], d[3]}, (v4f*)(out + base));
            __builtin_nontemporal_store((v4f){d[4], d[5], d[6], d[7]}, (v4f*)(out + base + 4));
        } else if (lane == 16) {
            __builtin_nontemporal_store((v4f){d[0], d[1], d[2], d[3]}, (v4f*)(out + base + 8));
            __builtin_nontemporal_store((v4f){d[4], d[5], d[6], d[7]}, (v4f*)(out + base + 12));
        }
    }
}

// Scalar fallback (only taken if F or S*F not divisible by 16).
__global__ void ar_finalize_scalar_kernel(const unsigned* __restrict__ mn,
                                          const unsigned* __restrict__ mx,
                                          const float* __restrict__ sums,
                                          const float* __restrict__ relu,
                                          const float* __restrict__ counts,
                                          const float* __restrict__ W,
                                          float* __restrict__ out,
                                          int SF, int F) {
    int e = blockIdx.x * blockDim.x + threadIdx.x;
    if (e >= SF) return;
    int s = e / F;
    out[e] = W[0] * counts[s] + W[1] * fdec(mn[e]) + W[2] * fdec(mx[e])
           + W[3] * relu[e]   + W[4] * sums[e];
}

extern "C" void kernel_launch(void* const* d_in, const int* in_sizes, int n_in,
                              void* d_out, int out_size, void* d_ws, size_t ws_size,
                              hipStream_t stream) {
    const float* x   = (const float*)d_in[0];
    const int*   idx = (const int*)d_in[1];
    // d_in[2] = max_index scalar (unused; derived from out_size)
    const float* t   = (const float*)d_in[3];
    const float* W   = (const float*)d_in[4];
    float* out = (float*)d_out;

    const int N  = in_sizes[1];
    const int F  = in_sizes[3];
    const int SF = out_size;
    const int S  = SF / F;

    // Workspace layout: mn_u | mx_u | sums | relu | counts  (~64.25 MB @ S=65536,F=64)
    unsigned* mn     = (unsigned*)d_ws;
    unsigned* mx     = mn + SF;
    float*    sums   = (float*)(mx + SF);
    float*    relu   = sums + SF;
    float*    counts = relu + SF;

    const int threads = 256;  // 8 wave32 waves per block

    ar_init_kernel<<<2048, threads, 0, stream>>>(mn, mx, sums, relu, counts, SF, S);

    const int  G       = F / 4;
    const long total   = (long)N * G;
    const int  blocksP = (int)((total + threads - 1) / threads);

    if (G == 16) {  // F == 64: shift/mask fast path, no integer divide
        ar_scatter_stats_kernel<4><<<blocksP, threads, 0, stream>>>(
            x, idx, mn, mx, sums, counts, N, F, G);
        ar_relu_scatter_kernel<4><<<blocksP, threads, 0, stream>>>(
            x, idx, t, mn, mx, relu, N, F, G);
    } else {
        ar_scatter_stats_kernel<-1><<<blocksP, threads, 0, stream>>>(
            x, idx, mn, mx, sums, counts, N, F, G);
        ar_relu_scatter_kernel<-1><<<blocksP, threads, 0, stream>>>(
            x, idx, t, mn, mx, relu, N, F, G);
    }

    if ((F % 16 == 0) && (SF % 16 == 0)) {
        const int ntiles  = SF / 16;                       // one wave per 16 outputs
        const int blocksF = (ntiles * 32 + threads - 1) / threads;
        ar_finalize_wmma_kernel<<<blocksF, threads, 0, stream>>>(
            mn, mx, sums, relu, counts, W, out, ntiles, F);
    } else {
        const int blocksF = (SF + threads - 1) / threads;
        ar_finalize_scalar_kernel<<<blocksF, threads, 0, stream>>>(
            mn, mx, sums, relu, counts, W, out, SF, F);
    }
}